// ViTClassifier_28449863369589
// MI455X (gfx1250) — compile-verified
//
#include <hip/hip_runtime.h>
#include <math.h>

// ---------------------------------------------------------------------------
// Model constants
// ---------------------------------------------------------------------------
#define S_TOK 4333
#define DMODEL 64
#define DFFN 512
#define MTILES ((S_TOK + 15) / 16)   // 271

typedef __attribute__((ext_vector_type(2))) float v2f;
typedef __attribute__((ext_vector_type(8))) float v8f;

// fp32 WMMA: D(16x16,f32) = A(16x4,f32) x B(4x16,f32) + C
__device__ __forceinline__ v8f wmma_f32(v2f a, v2f b, v8f c) {
  return __builtin_amdgcn_wmma_f32_16x16x4_f32(false, a, false, b, (short)0, c,
                                               false, false);
}

#define FLAG_ACC 1
#define FLAG_GELU 2

// ---------------------------------------------------------------------------
// Generic GEMM:  C[M,N] = A[M,K] @ W[N,K]^T + bias[N]   (torch Linear layout)
// flags: FLAG_ACC  -> C += ...,  FLAG_GELU -> exact gelu epilogue
// block = 128 threads (4 waves); each wave owns a 16x16 output tile.
// grid = (ceil(M/16), N/64).  N, K multiples of 16 (here 64/256/512).
// Out-of-range A rows are CLAMPED, not zeroed: they only affect out-of-range
// output rows, which are never stored.  Inner loop is branch-free:
// global_load_b64 x2 + v_wmma per k-step, with prefetch per 16-wide k block.
// ---------------------------------------------------------------------------
__global__ __launch_bounds__(128) void gemm_wmma_kernel(
    const float* __restrict__ A, const float* __restrict__ W,
    const float* __restrict__ bias, float* __restrict__ C, int M, int N, int K,
    int flags) {
  const int lane = threadIdx.x & 31;
  const int wave = threadIdx.x >> 5;
  const int m = lane & 15;    // A-row within tile / output column lane
  const int half = lane >> 4; // 0: k-pair {0,1}, 1: k-pair {2,3}
  const int row0 = blockIdx.x * 16;
  const int col0 = blockIdx.y * 64 + wave * 16;
  const int arow = row0 + m;
  const int arowc = (arow < M) ? arow : (M - 1); // clamp, don't zero
  const float* Ap = A + (size_t)arowc * K + 2 * half;
  const float* Wp = W + (size_t)(col0 + m) * K + 2 * half;

  v8f acc = {};
  const int ccol = col0 + m;
  const int crow0 = row0 + 8 * half; // C layout: vgpr r -> row crow0+r
  if (flags & FLAG_ACC) {
#pragma unroll
    for (int r = 0; r < 8; ++r) {
      int rr = crow0 + r;
      acc[r] = (rr < M) ? C[(size_t)rr * N + ccol] : 0.0f;
    }
  }

  for (int k0 = 0; k0 < K; k0 += 16) {
    __builtin_prefetch(Ap + k0 + 16, 0, 3);
    __builtin_prefetch(Wp + k0 + 16, 0, 3);
#pragma unroll
    for (int kk = 0; kk < 16; kk += 4) {
      v2f a = *(const v2f*)(Ap + k0 + kk);
      v2f b = *(const v2f*)(Wp + k0 + kk);
      acc = wmma_f32(a, b, acc);
    }
  }

  const float bv = bias[ccol];
#pragma unroll
  for (int r = 0; r < 8; ++r) {
    int rr = crow0 + r;
    if (rr < M) {
      float v = acc[r] + bv;
      if (flags & FLAG_GELU)
        v = 0.5f * v * (1.0f + erff(v * 0.70710678118654752f));
      C[(size_t)rr * N + ccol] = v;
    }
  }
}

// ---------------------------------------------------------------------------
// Flash attention, NH=2, DK=32, fp32 WMMA. One wave per (16-row Q tile, head).
// Never materializes the S x S score matrix.  All tile loads are clamped
// (branch-free); invalid key columns are killed by the -1e30 mask so their
// softmax weight is exactly 0, making clamped V rows harmless (0 * finite).
// ---------------------------------------------------------------------------
__global__ __launch_bounds__(32) void flash_attn_kernel(
    const float* __restrict__ Q, const float* __restrict__ Km,
    const float* __restrict__ Vm, float* __restrict__ O, int M) {
  __shared__ float Ps[16 * 16];
  const int lane = threadIdx.x;
  const int m = lane & 15;
  const int half = lane >> 4;
  const int h = blockIdx.y;
  const int row0 = blockIdx.x * 16;
  const int qrow = row0 + m;
  const int qrowc = (qrow < M) ? qrow : (M - 1);
  const float scale = 0.17677669529663687f; // 1/sqrt(32)
  const float* Qp = Q + (size_t)qrowc * DMODEL + h * 32 + 2 * half;

  // Q tile as A-fragments for 8 k-steps (DK=32); fold in softmax scale
  v2f qa[8];
#pragma unroll
  for (int kk = 0; kk < 8; ++kk) {
    v2f q = *(const v2f*)(Qp + 4 * kk);
    qa[kk].x = q.x * scale;
    qa[kk].y = q.y * scale;
  }

  float mrow[8], lrow[8];
  v8f o0 = {}, o1 = {};
#pragma unroll
  for (int r = 0; r < 8; ++r) { mrow[r] = -1e30f; lrow[r] = 0.0f; }

  const int ntiles = (M + 15) >> 4;
  for (int t = 0; t < ntiles; ++t) {
    const int krow = t * 16 + m;
    const int krowc = (krow < M) ? krow : (M - 1);
    const float colmask = (krow < M) ? 0.0f : -1e30f;
    const float* Kp = Km + (size_t)krowc * DMODEL + h * 32 + 2 * half;

    v8f s = {};
#pragma unroll
    for (int kk = 0; kk < 8; ++kk) {
      v2f b = *(const v2f*)(Kp + 4 * kk);
      s = wmma_f32(qa[kk], b, s);
    }

    // online softmax: rows live in 16-lane halves; reduce with shfl_xor < 16
#pragma unroll
    for (int r = 0; r < 8; ++r) {
      float v = s[r] + colmask;
      float rm = v;
#pragma unroll
      for (int off = 1; off < 16; off <<= 1)
        rm = fmaxf(rm, __shfl_xor(rm, off, 32));
      float newm = fmaxf(mrow[r], rm);
      float sc = __expf(mrow[r] - newm);
      float p = __expf(v - newm);
      float rs = p;
#pragma unroll
      for (int off = 1; off < 16; off <<= 1) rs += __shfl_xor(rs, off, 32);
      lrow[r] = lrow[r] * sc + rs;
      mrow[r] = newm;
      o0[r] *= sc;
      o1[r] *= sc;
      s[r] = p;
    }

    // C-layout -> A-layout for P via LDS round trip
#pragma unroll
    for (int r = 0; r < 8; ++r) Ps[(r + 8 * half) * 16 + m] = s[r];
    __syncthreads();
#pragma unroll
    for (int kk = 0; kk < 4; ++kk) { // K=16 in 4 steps
      const int kp = 4 * kk + 2 * half;
      v2f pa;
      pa.x = Ps[m * 16 + kp];
      pa.y = Ps[m * 16 + kp + 1];
      const int vr0 = t * 16 + kp;
      const int vr1 = vr0 + 1;
      const float* V0 = Vm + (size_t)(vr0 < M ? vr0 : M - 1) * DMODEL + h * 32;
      const float* V1 = Vm + (size_t)(vr1 < M ? vr1 : M - 1) * DMODEL + h * 32;
      v2f b0, b1;
      b0.x = V0[m];       // p==0 for masked cols -> clamped V rows contribute 0
      b0.y = V1[m];
      b1.x = V0[16 + m];
      b1.y = V1[16 + m];
      o0 = wmma_f32(pa, b0, o0);
      o1 = wmma_f32(pa, b1, o1);
    }
    __syncthreads();
  }

#pragma unroll
  for (int r = 0; r < 8; ++r) {
    int rr = row0 + r + 8 * half;
    if (rr < M) {
      float inv = 1.0f / lrow[r];
      O[(size_t)rr * DMODEL + h * 32 + m] = o0[r] * inv;
      O[(size_t)rr * DMODEL + h * 32 + 16 + m] = o1[r] * inv;
    }
  }
}

// ---------------------------------------------------------------------------
// LayerNorm over D=64, one row per 64-thread block.
// If Vh != nullptr: out = LN(U; gE,bE) * LN(Vh; gH,bH)   (E*H gating)
// else:             out = LN(U; gE,bE)
// ---------------------------------------------------------------------------
__device__ __forceinline__ float redsum64(float v, float* sm, int t) {
  sm[t] = v;
  __syncthreads();
#pragma unroll
  for (int st = 32; st > 0; st >>= 1) {
    if (t < st) sm[t] += sm[t + st];
    __syncthreads();
  }
  float r = sm[0];
  __syncthreads();
  return r;
}

__global__ __launch_bounds__(64) void ln_kernel(
    const float* __restrict__ U, const float* __restrict__ Vh,
    const float* __restrict__ gE, const float* __restrict__ bE,
    const float* __restrict__ gH, const float* __restrict__ bH,
    float* __restrict__ out) {
  __shared__ float sm[64];
  const int s = blockIdx.x, t = threadIdx.x;
  const size_t base = (size_t)s * DMODEL;
  float u = U[base + t];
  float mu = redsum64(u, sm, t) * (1.0f / 64.0f);
  float du = u - mu;
  float var = redsum64(du * du, sm, t) * (1.0f / 64.0f);
  float res = du * rsqrtf(var + 1e-5f) * gE[t] + bE[t];
  if (Vh) {
    float v = Vh[base + t];
    float mv = redsum64(v, sm, t) * (1.0f / 64.0f);
    float dv = v - mv;
    float vv = redsum64(dv * dv, sm, t) * (1.0f / 64.0f);
    res *= dv * rsqrtf(vv + 1e-5f) * gH[t] + bH[t];
  }
  out[base + t] = res;
}

// ---------------------------------------------------------------------------
// Elementwise kernels
// ---------------------------------------------------------------------------
__global__ void build_x_kernel(const float* __restrict__ x,
                               const float* __restrict__ cls,
                               const float* __restrict__ pos,
                               float* __restrict__ X, int n) {
  int i = blockIdx.x * blockDim.x + threadIdx.x;
  if (i < n) {
    float base = (i < DMODEL) ? cls[i] : x[i - DMODEL];
    X[i] = base + pos[i];
  }
}

__global__ void add_kernel(const float* __restrict__ a,
                           const float* __restrict__ b, float* __restrict__ o,
                           int n) {
  int i = blockIdx.x * blockDim.x + threadIdx.x;
  if (i < n) o[i] = a[i] + b[i];
}

// mLSTM gate combine: only Ht is needed downstream.
__global__ void lstm_combine_kernel(
    const float* __restrict__ Gi, const float* __restrict__ Gf,
    const float* __restrict__ Go, const float* __restrict__ Gz,
    const float* __restrict__ C, const float* __restrict__ M,
    const float* __restrict__ N, float* __restrict__ H, int n) {
  int i = blockIdx.x * blockDim.x + threadIdx.x;
  if (i < n) {
    float It = Gi[i], Ft = Gf[i], Ot = Go[i], Zt = Gz[i];
    float Mi = M[i];
    float Mt = fmaxf(Ft + Mi, It);
    float Ig = __expf(It - Mt);
    float Fg = __expf(Ft + Mi - Mt);
    float Og = 1.0f / (1.0f + __expf(-Ot));
    float Nt = Fg * N[i] + Ig;
    float Ct = C[i] * Fg + tanhf(Zt) * Ig;
    H[i] = Og * (Ct / Nt);
  }
}

// ---------------------------------------------------------------------------
// Input index map.  Assumes jax pytree flattening of `params` (dict keys
// sorted ASCII-alphabetically at every level; uppercase < lowercase).
// Top level: x,C1,M1,H1,N1,C2,M2,H2,N2, params{cls,l1,l2,pos,t1,t2}
// ---------------------------------------------------------------------------
enum {
  IN_X = 0, IN_C1, IN_M1, IN_H1, IN_N1, IN_C2, IN_M2, IN_H2, IN_N2,
  IN_CLS = 9,
  // l1: RF,RI,RO,RZ, WF,WI,WO,WZ, bRF,bRI,bRO,bRZ, bWF,bWI,bWO,bWZ
  L1_RF = 10, L1_RI, L1_RO, L1_RZ, L1_WF, L1_WI, L1_WO, L1_WZ,
  L1_bRF, L1_bRI, L1_bRO, L1_bRZ, L1_bWF, L1_bWI, L1_bWO, L1_bWZ,
  // l2: 26..41 (dead code in reference; unused)
  IN_POS = 42,
  T1_BASE = 43,
  T2_BASE = 73,
};
// tparams leaves, ASCII-sorted (30):
enum {
  T_W1 = 0, T_W2, T_WCk, T_WCq, T_WCv, T_Wk, T_Wq, T_Wv,
  T_b1, T_b1n, T_b2, T_bCk, T_bCq, T_bCv, T_bKE, T_bKH, T_bQE, T_bQH,
  T_bVE, T_bVH, T_bk, T_bq, T_bv,
  T_g1, T_gKE, T_gKH, T_gQE, T_gQH, T_gVE, T_gVH
};

extern "C" void kernel_launch(void* const* d_in, const int* in_sizes, int n_in,
                              void* d_out, int out_size, void* d_ws,
                              size_t ws_size, hipStream_t stream) {
  (void)in_sizes; (void)n_in; (void)out_size; (void)ws_size;
  auto in = [&](int i) { return (const float*)d_in[i]; };

  const int S = S_TOK;
  const size_t S64 = (size_t)S * DMODEL;
  const size_t S256 = (size_t)S * 256;
  const size_t S512 = (size_t)S * DFFN;

  float* ws = (float*)d_ws;
  float* X = ws;
  float* Ub = X + S64;
  float* Vb = Ub + S64;
  float* Qp = Vb + S64;
  float* Kp = Qp + S64;
  float* Vp = Kp + S64;
  float* Att = Vp + S64;
  float* ZA = Att + S64;   // state + att (intermediate)
  float* LN1 = ZA + S64;
  float* Z4 = LN1 + S64;
  float* ZOUT1 = Z4 + S64; // tblock1 output
  float* T1b = ZOUT1 + S64;   // [S,256]
  float* H1n = T1b + S256;    // [S,512]
  float* G0 = H1n + S512;     // I gate preact
  float* G1 = G0 + S512;      // F
  float* G2 = G1 + S512;      // O
  float* G3 = G2 + S512;      // Z

  auto gemm = [&](const float* A, const float* W, const float* bias, float* C,
                  int N, int K, int flags) {
    dim3 g(MTILES, N / 64);
    gemm_wmma_kernel<<<g, 128, 0, stream>>>(A, W, bias, C, S, N, K, flags);
  };
  auto eadd = [&](const float* a, const float* b, float* o, int n) {
    add_kernel<<<(n + 255) / 256, 256, 0, stream>>>(a, b, o, n);
  };

  // x = concat(cls, x) + pos
  build_x_kernel<<<((int)S64 + 255) / 256, 256, 0, stream>>>(
      in(IN_X), in(IN_CLS), in(IN_POS), X, (int)S64);

  // transformer block: state[S,64], Hs[S,512]; Zmid = state+att; Zout = Zmid+Z4
  auto tblock = [&](const float* state, const float* Hs, int TB, float* Zmid,
                    float* Zout) {
    auto T = [&](int o) { return in(TB + o); };
    // q
    gemm(state, T(T_Wq), T(T_bq), Ub, 64, 64, 0);
    gemm(Hs, T(T_WCq), T(T_bCq), Vb, 64, 512, 0);
    ln_kernel<<<S, 64, 0, stream>>>(Ub, Vb, T(T_gQE), T(T_bQE), T(T_gQH),
                                    T(T_bQH), Qp);
    // k
    gemm(state, T(T_Wk), T(T_bk), Ub, 64, 64, 0);
    gemm(Hs, T(T_WCk), T(T_bCk), Vb, 64, 512, 0);
    ln_kernel<<<S, 64, 0, stream>>>(Ub, Vb, T(T_gKE), T(T_bKE), T(T_gKH),
                                    T(T_bKH), Kp);
    // v
    gemm(state, T(T_Wv), T(T_bv), Ub, 64, 64, 0);
    gemm(Hs, T(T_WCv), T(T_bCv), Vb, 64, 512, 0);
    ln_kernel<<<S, 64, 0, stream>>>(Ub, Vb, T(T_gVE), T(T_bVE), T(T_gVH),
                                    T(T_bVH), Vp);
    // attention
    flash_attn_kernel<<<dim3(MTILES, 2), 32, 0, stream>>>(Qp, Kp, Vp, Att, S);
    // residual + FFN
    eadd(state, Att, Zmid, (int)S64);
    ln_kernel<<<S, 64, 0, stream>>>(Zmid, nullptr, T(T_g1), T(T_b1n), nullptr,
                                    nullptr, LN1);
    gemm(LN1, T(T_W1), T(T_b1), T1b, 256, 64, FLAG_GELU);
    gemm(T1b, T(T_W2), T(T_b2), Z4, 64, 256, 0);
    eadd(Zmid, Z4, Zout, (int)S64);
  };

  // ---- block 1 ----
  tblock(X, in(IN_H1), T1_BASE, ZA, ZOUT1);

  // ---- mLSTM 1: only H1n needed ----
  // gate = Hh @ W^T + bW  +  Z1 @ R^T + bR
  gemm(in(IN_H1), in(L1_WI), in(L1_bWI), G0, 512, 512, 0);
  gemm(ZOUT1, in(L1_RI), in(L1_bRI), G0, 512, 64, FLAG_ACC);
  gemm(in(IN_H1), in(L1_WF), in(L1_bWF), G1, 512, 512, 0);
  gemm(ZOUT1, in(L1_RF), in(L1_bRF), G1, 512, 64, FLAG_ACC);
  gemm(in(IN_H1), in(L1_WO), in(L1_bWO), G2, 512, 512, 0);
  gemm(ZOUT1, in(L1_RO), in(L1_bRO), G2, 512, 64, FLAG_ACC);
  gemm(in(IN_H1), in(L1_WZ), in(L1_bWZ), G3, 512, 512, 0);
  gemm(ZOUT1, in(L1_RZ), in(L1_bRZ), G3, 512, 64, FLAG_ACC);
  lstm_combine_kernel<<<((int)S512 + 255) / 256, 256, 0, stream>>>(
      G0, G1, G2, G3, in(IN_C1), in(IN_M1), in(IN_N1), H1n, (int)S512);

  // ---- block 2 (second _lstm's outputs are discarded by the reference) ----
  tblock(ZOUT1, H1n, T2_BASE, ZA, (float*)d_out);
}